// GCNClassifier_64209761075716
// MI455X (gfx1250) — compile-verified
//
#include <hip/hip_runtime.h>
#include <hip/hip_bf16.h>
#include <math.h>

// ---------------------------------------------------------------------------
// GCN classifier pipeline for MI455X (gfx1250, wave32).
//   deg -> xs = x*rsqrt(deg_out+1) -> scatter agg[dst]+=xs[src] (+self loop)
//   h = relu((agg*rsqrt(deg_in+1)) @ W + b)  [fp32 WMMA 16x16x4]
//   hg = segment_max(h)  [fused into GEMM epilogue via int-bits atomicMax,
//                         valid because relu output is non-negative]
//   out = log_softmax(hg @ Wc + bc)
// ---------------------------------------------------------------------------

typedef __attribute__((ext_vector_type(2))) float v2f;
typedef __attribute__((ext_vector_type(8))) float v8f;

#define IN_DIM 128
#define HID    256
#define NCLS   16
#define LDSP   132   // padded LDS row stride (floats): 132 mod 64 = 4 -> conflict-free

// ---------------- zero init ----------------
__global__ void zero_i32(int* __restrict__ p, int n) {
  int i = blockIdx.x * blockDim.x + threadIdx.x;
  if (i < n) p[i] = 0;
}

// ---------------- degree count ----------------
__global__ void degree_kernel(const int* __restrict__ src, const int* __restrict__ dst,
                              int* __restrict__ deg_out, int* __restrict__ deg_in, int nE) {
  int i = blockIdx.x * blockDim.x + threadIdx.x;
  int stride = gridDim.x * blockDim.x;
  for (int e = i; e < nE; e += stride) {
    atomicAdd(&deg_out[src[e]], 1);
    atomicAdd(&deg_in[dst[e]], 1);
  }
}

// ---------------- xs = x * rsqrt(deg_out+1); agg initialized to xs (self loop)
__global__ void scale_src_kernel(const float* __restrict__ x, const int* __restrict__ deg_out,
                                 float* __restrict__ xs, float* __restrict__ agg, int nNodes) {
  int t = blockIdx.x * blockDim.x + threadIdx.x;   // nNodes*32 threads, 4 floats each
  int node = t >> 5;
  if (node >= nNodes) return;
  int c4 = (t & 31) << 2;
  float s = rsqrtf((float)deg_out[node] + 1.0f);
  float4 v = *(const float4*)(x + (size_t)node * IN_DIM + c4);
  v.x *= s; v.y *= s; v.z *= s; v.w *= s;
  *(float4*)(xs  + (size_t)node * IN_DIM + c4) = v;
  *(float4*)(agg + (size_t)node * IN_DIM + c4) = v;
}

// ---------------- edge scatter: agg[dst] += xs[src]  (one wave per edge) ----
__global__ void scatter_kernel(const int* __restrict__ src, const int* __restrict__ dst,
                               const float* __restrict__ xs, float* __restrict__ agg, int nE) {
  int t = blockIdx.x * blockDim.x + threadIdx.x;
  int e = t >> 5;
  if (e >= nE) return;
  int c4 = (t & 31) << 2;                           // wave covers 128 cols contiguously
  int s = src[e], d = dst[e];
  float4 v = *(const float4*)(xs + (size_t)s * IN_DIM + c4);
  float* a = agg + (size_t)d * IN_DIM + c4;
  atomicAdd(a + 0, v.x); atomicAdd(a + 1, v.y);
  atomicAdd(a + 2, v.z); atomicAdd(a + 3, v.w);
}

// ---------------- fp32 WMMA GEMM + bias + relu + fused max-pool -------------
// grid = (ceil(M/16), HID/128), block = 256 (8 waves; wave w owns out cols
// [blockIdx.y*128 + w*16, +16)). A tile (16 x 128) staged in LDS, scaled by
// rsqrt(deg_in+1). K consumed in 32 steps of V_WMMA_F32_16X16X4_F32.
__global__ __launch_bounds__(256) void gemm_relu_pool_kernel(
    const float* __restrict__ agg, const int* __restrict__ deg_in,
    const float* __restrict__ W, const float* __restrict__ bias,
    const int* __restrict__ graph_ids, int* __restrict__ hg_bits, int nNodes)
{
  __shared__ float sA[16 * LDSP];
  const int rowBase = blockIdx.x * 16;
  const int colBase = blockIdx.y * 128;

  // cooperative A-tile load: thread t loads row t/16, 8 cols at (t%16)*8
  {
    int t  = threadIdx.x;
    int r  = t >> 4;
    int c8 = (t & 15) << 3;
    int row = rowBase + r;
    float4 v0 = {0.f, 0.f, 0.f, 0.f}, v1 = {0.f, 0.f, 0.f, 0.f};
    if (row < nNodes) {
      float s = rsqrtf((float)deg_in[row] + 1.0f);   // D_in^{-1/2}
      const float* p = agg + (size_t)row * IN_DIM + c8;
      v0 = *(const float4*)(p);
      v1 = *(const float4*)(p + 4);
      v0.x *= s; v0.y *= s; v0.z *= s; v0.w *= s;
      v1.x *= s; v1.y *= s; v1.z *= s; v1.w *= s;
    }
    *(float4*)&sA[r * LDSP + c8]     = v0;
    *(float4*)&sA[r * LDSP + c8 + 4] = v1;
  }
  __syncthreads();

  const int lane = threadIdx.x & 31;
  const int wave = threadIdx.x >> 5;        // 0..7
  const int hl   = lane >> 4;               // lane half: 0 or 1
  const int l16  = lane & 15;
  const int gn   = colBase + wave * 16 + l16;   // global output column

  v8f c = {};
  #pragma unroll
  for (int kk = 0; kk < IN_DIM / 4; ++kk) {
    // A 16x4 f32 layout: lanes 0-15 -> K = kb,kb+1 ; lanes 16-31 -> K = kb+2,kb+3
    int kb = kk * 4 + 2 * hl;
    const float* ap = &sA[l16 * LDSP + kb];
    v2f av; av.x = ap[0]; av.y = ap[1];
    // B 4x16 f32 layout mirrors: VGPR0/1 hold rows (kb, kb+1), cols = l16
    v2f bv; bv.x = W[(size_t)kb * HID + gn];
            bv.y = W[(size_t)(kb + 1) * HID + gn];
    c = __builtin_amdgcn_wmma_f32_16x16x4_f32(
            /*neg_a=*/false, av, /*neg_b=*/false, bv,
            /*c_mod=*/(short)0, c, /*reuse_a=*/false, /*reuse_b=*/false);
  }

  // epilogue: bias + relu, fused per-graph max pool.
  // C/D layout: VGPR i -> M = i (lanes 0-15) or i+8 (lanes 16-31), N = l16.
  float bn = bias[gn];
  #pragma unroll
  for (int i = 0; i < 8; ++i) {
    int row = rowBase + i + 8 * hl;
    if (row < nNodes) {
      float v = c[i] + bn;
      v = v > 0.0f ? v : 0.0f;                 // relu -> v >= 0
      int g = graph_ids[row];
      // non-negative IEEE floats order like their int bit patterns
      atomicMax(&hg_bits[g * HID + gn], __float_as_int(v));
    }
  }
}

// ---------------- classifier head: logits + log_softmax ---------------------
__global__ void head_kernel(const int* __restrict__ hg_bits, const float* __restrict__ Wc,
                            const float* __restrict__ bc, float* __restrict__ out, int nGraphs) {
  int g = blockIdx.x * blockDim.x + threadIdx.x;
  if (g >= nGraphs) return;
  const float* hg = (const float*)hg_bits + (size_t)g * HID;
  float logits[NCLS];
  #pragma unroll
  for (int cc = 0; cc < NCLS; ++cc) logits[cc] = bc[cc];
  for (int k = 0; k < HID; ++k) {
    float hv = hg[k];
    const float* wr = Wc + (size_t)k * NCLS;
    #pragma unroll
    for (int cc = 0; cc < NCLS; ++cc) logits[cc] += hv * wr[cc];
  }
  float m = logits[0];
  #pragma unroll
  for (int cc = 1; cc < NCLS; ++cc) m = fmaxf(m, logits[cc]);
  float s = 0.0f;
  #pragma unroll
  for (int cc = 0; cc < NCLS; ++cc) s += expf(logits[cc] - m);
  float lse = logf(s) + m;
  #pragma unroll
  for (int cc = 0; cc < NCLS; ++cc) out[(size_t)g * NCLS + cc] = logits[cc] - lse;
}

// ---------------------------------------------------------------------------
extern "C" void kernel_launch(void* const* d_in, const int* in_sizes, int n_in,
                              void* d_out, int out_size, void* d_ws, size_t ws_size,
                              hipStream_t stream) {
  const float* x  = (const float*)d_in[0];
  const int*   sr = (const int*)d_in[1];
  const int*   ds = (const int*)d_in[2];
  const int*   gi = (const int*)d_in[3];
  const float* W  = (const float*)d_in[4];
  const float* b  = (const float*)d_in[5];
  const float* Wc = (const float*)d_in[6];
  const float* bc = (const float*)d_in[7];

  const int nNodes  = in_sizes[3];
  const int nE      = in_sizes[1];
  const int nGraphs = out_size / NCLS;

  // workspace carve-out (all 16B-aligned: nNodes*4 = 400000 is a mult of 16)
  char* ws = (char*)d_ws;
  int*   deg_out = (int*)ws;   ws += (size_t)nNodes * sizeof(int);
  int*   deg_in  = (int*)ws;   ws += (size_t)nNodes * sizeof(int);
  float* xs      = (float*)ws; ws += (size_t)nNodes * IN_DIM * sizeof(float);
  float* agg     = (float*)ws; ws += (size_t)nNodes * IN_DIM * sizeof(float);
  int*   hg_bits = (int*)ws;   // nGraphs * HID ints (float bit patterns)

  const int T = 256;

  zero_i32<<<(2 * nNodes + T - 1) / T, T, 0, stream>>>(deg_out, 2 * nNodes);
  zero_i32<<<(nGraphs * HID + T - 1) / T, T, 0, stream>>>(hg_bits, nGraphs * HID);

  degree_kernel<<<1024, T, 0, stream>>>(sr, ds, deg_out, deg_in, nE);

  {
    long long tot = (long long)nNodes * 32;
    scale_src_kernel<<<(unsigned)((tot + T - 1) / T), T, 0, stream>>>(x, deg_out, xs, agg, nNodes);
  }
  {
    long long tot = (long long)nE * 32;
    scatter_kernel<<<(unsigned)((tot + T - 1) / T), T, 0, stream>>>(sr, ds, xs, agg, nE);
  }

  dim3 gg((nNodes + 15) / 16, HID / 128);
  gemm_relu_pool_kernel<<<gg, 256, 0, stream>>>(agg, deg_in, W, b, gi, hg_bits, nNodes);

  head_kernel<<<(nGraphs + 63) / 64, 64, 0, stream>>>(hg_bits, Wc, bc, (float*)d_out, nGraphs);
}